// MonotoneActivation_58394375357254
// MI455X (gfx1250) — compile-verified
//
#include <hip/hip_runtime.h>
#include <hip/hip_bf16.h>
#include <stdint.h>

// MonotoneActivation (simplex-interpolation lattice activation), MI455X/gfx1250.
// Memory-bound (AI ~1.4 FLOP/B): X in 32MB + out 64MB + params 256KB @ 23.3 TB/s -> ~4.1us floor.
// Strategy: stage params tables into LDS once per block via gfx1250 async-to-LDS DMA
// (ASYNCcnt path), then loop over 16 batch tiles: branchless 4-element sorting network,
// data-dependent LDS gathers, non-temporal streaming loads/stores for X and out.

typedef float v4f __attribute__((ext_vector_type(4)));

#define G_TILE  64
#define B_TILE  4
#define B_ITER  16      // batch tiles per block: 64 rows per block
#define BLOCK   256
#define TBL_PAD 136     // 128 floats of table + 8 pad floats (544B, 32B-aligned, bank spread)

// min/max based compare-swap carrying one-hot index masks
#define CSWAP(va, ma, vb, mb) do {                    \
    const bool  _sw = (va) > (vb);                    \
    const float _lo = fminf((va), (vb));              \
    const float _hi = fmaxf((va), (vb));              \
    (va) = _lo; (vb) = _hi;                           \
    const int   _mt = _sw ? (ma) : (mb);              \
    (ma) = _sw ? (mb) : (ma); (mb) = _mt;             \
} while (0)

__global__ __launch_bounds__(BLOCK)
void MonotoneActivation_58394375357254_kernel(const float* __restrict__ X,
                                              const float* __restrict__ params,
                                              float* __restrict__ out,
                                              int G) {
    __shared__ float sP[G_TILE * TBL_PAD];   // 64 tables of 16x8 floats, padded

    const int tid = threadIdx.x;
    const int g0  = blockIdx.x * G_TILE;               // first group of this block
    const int b0  = blockIdx.y * (B_TILE * B_ITER);    // first batch row of this block

    // ---- Stage 64 param tables (32KB) into LDS with gfx1250 async DMA, once per block ----
    {
        const v4f* pbase = (const v4f*)params + (size_t)g0 * 32;
        #pragma unroll
        for (int j = tid; j < G_TILE * 32; j += BLOCK) {
            const int t = j >> 5;            // table within tile
            const int c = j & 31;            // float4 chunk within table
            unsigned loff = (unsigned)(uintptr_t)&sP[t * TBL_PAD + (c << 2)];
            unsigned long long ga = (unsigned long long)(uintptr_t)(pbase + (t << 5) + c);
            asm volatile("global_load_async_to_lds_b128 %0, %1, off"
                         :: "v"(loff), "v"(ga) : "memory");
        }
    }

    const int gl = tid & (G_TILE - 1);       // group within tile
    const int bl = tid >> 6;                 // batch row within tile
    const int g  = g0 + gl;

    const size_t xstride = (size_t)G * 4;    // floats per X row
    const size_t ostride = (size_t)G * 8;    // floats per out row
    const float* tb = &sP[gl * TBL_PAD];

    // ---- Wait for async DMA, publish LDS to all waves ----
    asm volatile("s_wait_asynccnt 0x0" ::: "memory");
    __syncthreads();

    // ---- Stream 16 batch tiles through the staged tables ----
    #pragma unroll 4
    for (int it = 0; it < B_ITER; ++it) {
        const int b = b0 + it * B_TILE + bl;

        const v4f* xrow = (const v4f*)(X + (size_t)b * xstride);
        const v4f x4 = __builtin_nontemporal_load(&xrow[g]);

        // Branchless 4-element sorting network carrying one-hot masks
        float v0 = x4.x, v1 = x4.y, v2 = x4.z, v3 = x4.w;
        int   m0 = 1,    m1 = 2,    m2 = 4,    m3 = 8;
        CSWAP(v0, m0, v1, m1);
        CSWAP(v2, m2, v3, m3);
        CSWAP(v0, m0, v2, m2);
        CSWAP(v1, m1, v3, m3);
        CSWAP(v1, m1, v2, m2);
        // Ascending; masks = 2^(original index) per rank. Suffix-sum table indices:
        const int i1 = 15 - m0;
        const int i2 = i1 - m1;
        const int i3 = m3;
        const float c0 = v0;
        const float c1 = v1 - v0;
        const float c2 = v2 - v1;
        const float c3 = v3 - v2;

        // Data-dependent gathers from LDS + weighted sum (D=8)
        const v4f* r0 = (const v4f*)(tb + (15 << 3));   // i0 == 15 always
        const v4f* r1 = (const v4f*)(tb + (i1 << 3));
        const v4f* r2 = (const v4f*)(tb + (i2 << 3));
        const v4f* r3 = (const v4f*)(tb + (i3 << 3));

        v4f a0 = c0 * r0[0];
        v4f a1 = c0 * r0[1];
        a0 += c1 * r1[0];  a1 += c1 * r1[1];
        a0 += c2 * r2[0];  a1 += c2 * r2[1];
        a0 += c3 * r3[0];  a1 += c3 * r3[1];

        // Coalesced streaming store: out[b, g*8 .. g*8+7]
        v4f* orow = (v4f*)(out + (size_t)b * ostride + (size_t)g * 8);
        __builtin_nontemporal_store(a0, &orow[0]);
        __builtin_nontemporal_store(a1, &orow[1]);
    }
}

extern "C" void kernel_launch(void* const* d_in, const int* in_sizes, int n_in,
                              void* d_out, int out_size, void* d_ws, size_t ws_size,
                              hipStream_t stream) {
    const float* X      = (const float*)d_in[0];   // (B, G*4) f32
    const float* params = (const float*)d_in[1];   // (G, 16, 8) f32
    float*       out    = (float*)d_out;           // (B, G*8) f32

    const int G = in_sizes[1] / (16 * 8);          // 512
    const int B = in_sizes[0] / (G * 4);           // 4096

    dim3 grid(G / G_TILE, B / (B_TILE * B_ITER));  // (8, 64)
    dim3 block(BLOCK);
    MonotoneActivation_58394375357254_kernel<<<grid, block, 0, stream>>>(X, params, out, G);
}